// RoutingByAgreement_48137993453828
// MI455X (gfx1250) — compile-verified
//
#include <hip/hip_runtime.h>
#include <hip/hip_bf16.h>

// Capsule routing-by-agreement, fused single-pass-per-iteration formulation.
// batch=128, ic=2048, oc=32, od=16, 3 routing iterations.
//
// Per workgroup (one batch element):
//   LDS: b_batch (2048x32 f32 = 256KB, CDNA5 320KB LDS), wave partials Sw,
//        s and v vectors.
//   4 streaming passes over u[b] (4MB each): pass0 uses softmax(b_prior);
//   iters 1..3 fuse agreement(u,v) + logit update + softmax + weighted sum.
// Cross-wave s-reduction done with V_WMMA_F32_16X16X4_F32 (ones-matrix GEMM).

typedef float v2f __attribute__((ext_vector_type(2)));
typedef float v8f __attribute__((ext_vector_type(8)));

#define IC 2048
#define OC 32
#define OD 16
#define WAVES 16
#define THREADS (WAVES * 32)
#define ROWS_PER_WAVE (IC / WAVES)
#define ROW_F (OC * OD) // 512 floats per (b,i) row, contiguous in memory

#if __has_builtin(__builtin_amdgcn_wmma_f32_16x16x4_f32)
#define HAVE_WMMA_F32X4 1
#else
#define HAVE_WMMA_F32X4 0
#endif

union U16 {
  float4 v[4];
  float f[16];
};

// softmax weight for this lane's logit across the 32 lanes of the wave
__device__ __forceinline__ float softmax32(float x) {
  float m = x;
#pragma unroll
  for (int off = 16; off > 0; off >>= 1) m = fmaxf(m, __shfl_xor(m, off, 32));
  float e = __expf(x - m);
  float s = e;
#pragma unroll
  for (int off = 16; off > 0; off >>= 1) s += __shfl_xor(s, off, 32);
  return e / s;
}

// One fused streaming pass over u[b]. FIRST: logits come from the global
// prior b (and initialize LDS b_batch). Otherwise: agreement with v is added
// to LDS b_batch first. Produces this wave's partial s in Sw[w].
template <bool FIRST>
__device__ __forceinline__ void stream_pass(const float* __restrict__ ub,
                                            const float* __restrict__ bprior,
                                            float* bb, float* Sw,
                                            const float* vv, int w, int lane) {
  float sacc[OD];
#pragma unroll
  for (int d = 0; d < OD; ++d) sacc[d] = 0.0f;

  // prime the pipeline with row i = w (lane l reads u[i, o=l, 0:16])
  U16 cur, nxt;
  {
    const float4* p =
        reinterpret_cast<const float4*>(ub + (size_t)w * ROW_F + lane * OD);
    cur.v[0] = p[0]; cur.v[1] = p[1]; cur.v[2] = p[2]; cur.v[3] = p[3];
  }
  float bcur = 0.0f;
  if constexpr (FIRST) bcur = bprior[w * OC + lane];

  for (int t = 0; t < ROWS_PER_WAVE; ++t) {
    const int i = t * WAVES + w;

    // prefetch next row before the dependent math (latency hiding)
    nxt = cur;
    float bnxt = bcur;
    if (t + 1 < ROWS_PER_WAVE) {
      const int in = (t + 1) * WAVES + w;
      const float4* q =
          reinterpret_cast<const float4*>(ub + (size_t)in * ROW_F + lane * OD);
      nxt.v[0] = q[0]; nxt.v[1] = q[1]; nxt.v[2] = q[2]; nxt.v[3] = q[3];
      if constexpr (FIRST) bnxt = bprior[in * OC + lane];
    }

    float bbv;
    if constexpr (FIRST) {
      bbv = bcur; // initialize b_batch from prior
    } else {
      float a = 0.0f; // agreement: dot(u[i,o,:], v[o,:])
#pragma unroll
      for (int d = 0; d < OD; ++d) a = fmaf(cur.f[d], vv[d], a);
      bbv = bb[i * OC + lane] + a;
    }
    bb[i * OC + lane] = bbv;

    const float c = softmax32(bbv); // coupling coefficient over oc
#pragma unroll
    for (int d = 0; d < OD; ++d) sacc[d] = fmaf(c, cur.f[d], sacc[d]);

    cur = nxt;
    bcur = bnxt;
  }

  // publish this wave's partial s[o=lane, 0:16]
  U16 so;
#pragma unroll
  for (int d = 0; d < OD; ++d) so.f[d] = sacc[d];
  float4* swp = reinterpret_cast<float4*>(Sw + w * (OC * OD) + lane * OD);
  swp[0] = so.v[0]; swp[1] = so.v[1]; swp[2] = so.v[2]; swp[3] = so.v[3];
}

// s_total[n] = sum over 16 wave-partials of Sw[k][n], n in [0,512).
// Expressed as D = ONES(16x4) x B(4x16) accumulated over 4 K-steps per
// 16-column chunk; A = all-ones is layout-invariant, B/C need only the
// documented lane&15 <-> N mapping. Each wave handles 2 of the 32 chunks.
__device__ __forceinline__ void reduce_partials(const float* Sw, float* sS,
                                                int w, int lane) {
#if HAVE_WMMA_F32X4
  v2f a;
  a[0] = 1.0f;
  a[1] = 1.0f;
#pragma unroll
  for (int j = 0; j < 2; ++j) {
    const int n0 = (w * 2 + j) * 16;
    const int col = n0 + (lane & 15);
    const int krow = (lane >> 4) << 1; // lanes 16-31 carry the other K pair
    v8f acc = {0.f, 0.f, 0.f, 0.f, 0.f, 0.f, 0.f, 0.f};
#pragma unroll
    for (int step = 0; step < 4; ++step) {
      const int k0 = step * 4 + krow;
      v2f bm;
      bm[0] = Sw[(k0 + 0) * (OC * OD) + col];
      bm[1] = Sw[(k0 + 1) * (OC * OD) + col];
      acc = __builtin_amdgcn_wmma_f32_16x16x4_f32(false, a, false, bm,
                                                  (short)0, acc, false, false);
    }
    // row M=0 of D = column sums; lanes 0-15 hold N = lane
    if (lane < 16) sS[n0 + lane] = acc[0];
  }
#else
  const int tid = w * 32 + lane;
  float acc = 0.0f;
#pragma unroll
  for (int k = 0; k < WAVES; ++k) acc += Sw[k * (OC * OD) + tid];
  sS[tid] = acc;
#endif
}

// v = s * sqrt(|s|^2) / (1 + |s|^2); wave 0, lane = o. Optionally emits output.
__device__ __forceinline__ void squash_store(const float* sS, float* vS,
                                             float* outp, int w, int lane) {
  if (w == 0) {
    float sv[OD];
    float sq = 0.0f;
#pragma unroll
    for (int d = 0; d < OD; ++d) {
      sv[d] = sS[lane * OD + d];
      sq = fmaf(sv[d], sv[d], sq);
    }
    const float scale = sqrtf(sq) / (1.0f + sq);
#pragma unroll
    for (int d = 0; d < OD; ++d) {
      const float vo = sv[d] * scale;
      vS[lane * OD + d] = vo;
      if (outp) outp[lane * OD + d] = vo;
    }
  }
}

__global__ __launch_bounds__(THREADS, 1) void
RoutingByAgreement_48137993453828_kernel(const float* __restrict__ u,
                                         const float* __restrict__ bprior,
                                         float* __restrict__ out) {
  extern __shared__ float smem[];
  float* bb = smem;                  // IC*OC        = 256 KB (b_batch)
  float* Sw = bb + IC * OC;          // WAVES*OC*OD  =  32 KB (wave partials)
  float* sS = Sw + WAVES * OC * OD;  // OC*OD        =   2 KB (reduced s)
  float* vS = sS + OC * OD;          // OC*OD        =   2 KB (v)

  const int w = threadIdx.x >> 5;
  const int lane = threadIdx.x & 31;
  const int b = blockIdx.x;
  const float* ub = u + (size_t)b * IC * ROW_F;

  // pass 0: c = softmax(prior), s = sum_i c*u ; init b_batch in LDS
  stream_pass<true>(ub, bprior, bb, Sw, nullptr, w, lane);
  __syncthreads();
  reduce_partials(Sw, sS, w, lane);
  __syncthreads();
  squash_store(sS, vS, nullptr, w, lane);
  __syncthreads();

#pragma unroll 1
  for (int it = 0; it < 3; ++it) {
    float vv[OD]; // v[o=lane, :] into registers for the fused pass
#pragma unroll
    for (int d = 0; d < OD; ++d) vv[d] = vS[lane * OD + d];
    stream_pass<false>(ub, nullptr, bb, Sw, vv, w, lane);
    __syncthreads();
    reduce_partials(Sw, sS, w, lane);
    __syncthreads();
    squash_store(sS, vS, (it == 2) ? (out + (size_t)b * OC * OD) : nullptr, w,
                 lane);
    __syncthreads();
  }
}

extern "C" void kernel_launch(void* const* d_in, const int* in_sizes, int n_in,
                              void* d_out, int out_size, void* d_ws,
                              size_t ws_size, hipStream_t stream) {
  const float* u = (const float*)d_in[0];
  const float* bp = (const float*)d_in[1];
  float* out = (float*)d_out;
  const int batch = in_sizes[0] / (IC * OC * OD);
  const size_t shmem =
      (size_t)(IC * OC + WAVES * OC * OD + 2 * OC * OD) * sizeof(float);
  (void)hipFuncSetAttribute(
      reinterpret_cast<const void*>(&RoutingByAgreement_48137993453828_kernel),
      hipFuncAttributeMaxDynamicSharedMemorySize, (int)shmem);
  RoutingByAgreement_48137993453828_kernel<<<batch, THREADS, shmem, stream>>>(
      u, bp, out);
}